// GlitchSampler_29162827940108
// MI455X (gfx1250) — compile-verified
//
#include <hip/hip_runtime.h>

// ---------------------------------------------------------------------------
// GlitchSampler for MI455X (gfx1250).
// Pure data movement: ~32-40 MB of traffic, 0 FLOPs -> HBM-bound (~1.7us at
// 23.3 TB/s). No matrix math, so WMMA does not apply; the CDNA5 feature that
// matches this op is the ASYNCcnt global<->LDS data-mover path
// (GLOBAL_LOAD_ASYNC_TO_LDS_* / GLOBAL_STORE_ASYNC_FROM_LDS_* + s_wait_asynccnt).
// ---------------------------------------------------------------------------

#define B_ 512
#define C_ 2
#define K_ 4096
#define NG_ 2048
#define T_ 16384
#define MIN_START_ 3584
#define TPB 256                   // 8 wave32s per block
#define CHUNKS (K_ / (TPB * 4))   // 4 float4 chunks per thread

typedef float f4 __attribute__((ext_vector_type(4)));
typedef int   v4i __attribute__((vector_size(4 * sizeof(int))));  // matches builtin param

typedef __attribute__((address_space(1))) void gvoid_t;
typedef __attribute__((address_space(3))) void lvoid_t;
typedef __attribute__((address_space(1))) int  gint_t;
typedef __attribute__((address_space(3))) int  lint_t;
typedef __attribute__((address_space(1))) v4i  gv4i_t;
typedef __attribute__((address_space(3))) v4i  lv4i_t;

// Device-pass-only detection of the gfx1250 async global<->LDS builtins.
#if defined(__AMDGCN__) &&                                                    \
    __has_builtin(__builtin_amdgcn_global_load_async_to_lds_b128) &&          \
    __has_builtin(__builtin_amdgcn_global_load_async_to_lds_b32)  &&          \
    __has_builtin(__builtin_amdgcn_global_store_async_from_lds_b128) &&       \
    __has_builtin(__builtin_amdgcn_s_wait_asynccnt)
#define USE_ASYNC_LDS 1
#else
#define USE_ASYNC_LDS 0
#endif

#if USE_ASYNC_LDS
// Cast helpers: generic -> AS(1)/AS(3) with the exact pointee types the
// builtins expect (as revealed by the round-1 diagnostics).
__device__ __forceinline__ gint_t* g32(const float* p) {
    return (gint_t*)(gvoid_t*)(void*)p;
}
__device__ __forceinline__ gv4i_t* g128(const float* p) {
    return (gv4i_t*)(gvoid_t*)(void*)p;
}
__device__ __forceinline__ lint_t* l32(float* p) {
    return (lint_t*)(lvoid_t*)p;
}
__device__ __forceinline__ lv4i_t* l128(float* p) {
    return (lv4i_t*)(lvoid_t*)p;
}
#endif

__global__ __launch_bounds__(TPB) void glitch_copy_kernel(
    const float* __restrict__ X,       // [B_, C_, K_]
    const float* __restrict__ gh,      // [NG_, T_]
    const float* __restrict__ gl,      // [NG_, T_]
    const int*   __restrict__ masks,   // [C_, B_]  (0 / nonzero)
    const int*   __restrict__ idx,     // [C_, B_]
    const int*   __restrict__ starts,  // [C_, B_]
    float* __restrict__ outX)          // [B_, C_, K_]
{
    const int r = blockIdx.x;          // row in [0, B_*C_)
    const int b = r >> 1;
    const int c = r & 1;
    const int t = threadIdx.x;

    const int m = masks[c * B_ + b];
    float* __restrict__ dst = outX + (size_t)(b * C_ + c) * K_;

    const float* __restrict__ src;
    if (m) {
        const float* __restrict__ g = (c == 0) ? gh : gl;
        const int gi = idx[c * B_ + b];
        const int s  = starts[c * B_ + b] + MIN_START_;
        src = g + (size_t)gi * T_ + s;       // only 4-byte aligned
    } else {
        src = X + (size_t)(b * C_ + c) * K_; // 16KB aligned row
    }

#if USE_ASYNC_LDS
    // ---- CDNA5 async data-mover path: global -> LDS -> global. Per-lane
    //      private LDS regions, so a per-wave s_wait_asynccnt(0) is the only
    //      synchronization needed between the two phases (no barrier).
    __shared__ float smem[K_];          // 16 KB of the 320 KB/WGP LDS

    if (m) {
        // Gather branch: source only 4B-aligned -> b32 async copies.
        #pragma unroll
        for (int j = 0; j < CHUNKS; ++j) {
            const int base = (j * TPB + t) * 4;
            #pragma unroll
            for (int q = 0; q < 4; ++q) {
                __builtin_amdgcn_global_load_async_to_lds_b32(
                    g32(src + base + q), l32(smem + base + q),
                    /*offset=*/0, /*cpol=*/0);
            }
        }
    } else {
        // Pass-through branch: 16B-aligned -> b128 async copies.
        #pragma unroll
        for (int j = 0; j < CHUNKS; ++j) {
            const int base = (j * TPB + t) * 4;
            __builtin_amdgcn_global_load_async_to_lds_b128(
                g128(src + base), l128(smem + base),
                /*offset=*/0, /*cpol=*/0);
        }
    }

    __builtin_amdgcn_s_wait_asynccnt(0);   // own wave's loads landed in LDS

    #pragma unroll
    for (int j = 0; j < CHUNKS; ++j) {
        const int base = (j * TPB + t) * 4;
        __builtin_amdgcn_global_store_async_from_lds_b128(
            g128(dst + base), l128(smem + base),
            /*offset=*/0, /*cpol=*/0);
    }
    __builtin_amdgcn_s_wait_asynccnt(0);   // drain before wave end
#else
    // ---- Fallback: direct streaming copy, B128 + non-temporal hints.
    if (m) {
        #pragma unroll
        for (int j = 0; j < CHUNKS; ++j) {
            const int base = (j * TPB + t) * 4;
            f4 v;
            v.x = __builtin_nontemporal_load(src + base + 0);
            v.y = __builtin_nontemporal_load(src + base + 1);
            v.z = __builtin_nontemporal_load(src + base + 2);
            v.w = __builtin_nontemporal_load(src + base + 3);
            __builtin_nontemporal_store(v, (f4*)(dst + base));
        }
    } else {
        const f4* __restrict__ s4 = (const f4*)src;
        #pragma unroll
        for (int j = 0; j < CHUNKS; ++j) {
            const int e = j * TPB + t;
            f4 v = __builtin_nontemporal_load(s4 + e);
            __builtin_nontemporal_store(v, (f4*)dst + e);
        }
    }
#endif
}

__global__ __launch_bounds__(TPB) void glitch_y_kernel(
    const float* __restrict__ y,
    const int*   __restrict__ masks,
    float* __restrict__ outY)
{
    const int b = blockIdx.x * blockDim.x + threadIdx.x;
    if (b < B_) {
        float v = y[b];
        if (masks[0 * B_ + b]) v -= 2.0f;
        if (masks[1 * B_ + b]) v -= 4.0f;
        outY[b] = v;
    }
}

extern "C" void kernel_launch(void* const* d_in, const int* in_sizes, int n_in,
                              void* d_out, int out_size, void* d_ws, size_t ws_size,
                              hipStream_t stream) {
    (void)in_sizes; (void)n_in; (void)out_size; (void)d_ws; (void)ws_size;

    const float* X      = (const float*)d_in[0];  // [B,C,K]
    const float* y      = (const float*)d_in[1];  // [B]
    const float* gh     = (const float*)d_in[2];  // [NG,T]
    const float* gl     = (const float*)d_in[3];  // [NG,T]
    const int*   masks  = (const int*)d_in[4];    // [C,B] bool -> int per harness
    const int*   idx    = (const int*)d_in[5];    // [C,B]
    const int*   starts = (const int*)d_in[6];    // [C,B]

    float* outX = (float*)d_out;                       // [B,C,K]
    float* outY = outX + (size_t)B_ * C_ * K_;         // [B]

    glitch_copy_kernel<<<B_ * C_, TPB, 0, stream>>>(X, gh, gl, masks, idx,
                                                    starts, outX);
    glitch_y_kernel<<<(B_ + TPB - 1) / TPB, TPB, 0, stream>>>(y, masks, outY);
}